// ElementwiseTensorProd_o0_9062380994709
// MI455X (gfx1250) — compile-verified
//
#include <hip/hip_runtime.h>

// ElementwiseTensorProd_o0 for MI455X (gfx1250).
// Memory-bound streaming kernel: AI ~0.16 FLOP/byte -> no WMMA (would be pure waste).
// z1 traffic (3/4 of load bytes) moves via the CDNA5 async global->LDS data mover,
// double-buffered per wave so S_WAIT_ASYNCCNT<=6 only ever waits on a batch that
// has had a full iteration of overlap. z0/out use non-temporal B128 accesses.

typedef float v4f __attribute__((ext_vector_type(4)));

#define RANK 128
#define Z1_BYTES_PER_TOKEN (3 * RANK * 4)      // 1536 B
#define WAVES_PER_BLOCK 8
#define LDS_PER_WAVE (4 * Z1_BYTES_PER_TOKEN)  // 6144 B: 2 buffers x (z1l + z1r)

// CDNA5: GLOBAL_LOAD_ASYNC_TO_LDS_B128, GVS mode, non-temporal global read.
//   reads from the 64-bit base in sbase plus the 32-bit offset in goff; writes to the LDS address in lds_addr
// Tracked by ASYNCcnt; async loads complete IN ORDER (ISA ch.8 "Async Memory" 4.1).
__device__ __forceinline__ void async_ld16(unsigned lds_addr, const char* sbase, unsigned goff) {
  asm volatile("global_load_async_to_lds_b128 %0, %1, %2 th:TH_LOAD_NT"
               :
               : "v"(lds_addr), "v"(goff), "s"(sbase)
               : "memory");
}

__device__ __forceinline__ void wait_async_le6() {
  // In-order completion => ASYNCcnt<=6 means every batch before the newest 6 landed.
  asm volatile("s_wait_asynccnt 6" ::: "memory");
}

// Stage one token of z1l+z1r (2 x 1536 B) into an LDS slab: 6 async B128s (512 B each).
__device__ __forceinline__ void stage_token(unsigned ldsA, unsigned ldsB,
                                            const char* z1l, const char* z1r,
                                            int t, int lane) {
  const unsigned gbase = (unsigned)t * (unsigned)Z1_BYTES_PER_TOKEN;  // < 2^29, fits
#pragma unroll
  for (int j = 0; j < 3; ++j) {
    const unsigned off = (unsigned)(j * 512 + lane * 16);
    async_ld16(ldsA + off, z1l, gbase + off);
    async_ld16(ldsB + off, z1r, gbase + off);
  }
}

__global__ __launch_bounds__(256) void etp_o0_kernel(
    const float* __restrict__ z0l, const float* __restrict__ z1l,
    const float* __restrict__ z0r, const float* __restrict__ z1r,
    float* __restrict__ out, int ntokens)
{
  __shared__ __align__(16) char lds[WAVES_PER_BLOCK * LDS_PER_WAVE];

  const int lane  = threadIdx.x & 31;            // wave32
  const int wave  = threadIdx.x >> 5;
  const int gwave = blockIdx.x * WAVES_PER_BLOCK + wave;
  const int nwave = gridDim.x * WAVES_PER_BLOCK;

  // Wave-private LDS: buf{0,1}, each buf = [z1l 1536B][z1r 1536B].
  char* base = lds + wave * LDS_PER_WAVE;
  const unsigned bufA[2] = { (unsigned)(size_t)base,
                             (unsigned)(size_t)(base + 2 * Z1_BYTES_PER_TOKEN) };

  int t = gwave;
  if (t >= ntokens) return;

  // Prologue: fill buffer 0 with token t.
  stage_token(bufA[0], bufA[0] + Z1_BYTES_PER_TOKEN,
              (const char*)z1l, (const char*)z1r, t, lane);

  int parity = 0;
  for (; t < ntokens; t += nwave, parity ^= 1) {
    // ---- prefetch next token into the other buffer (clamped so the in-order
    // ASYNCcnt<=6 wait below stays correct on the final iteration).
    const int tn = (t + nwave < ntokens) ? (t + nwave) : t;
    const unsigned nb = bufA[parity ^ 1];
    stage_token(nb, nb + Z1_BYTES_PER_TOKEN,
                (const char*)z1l, (const char*)z1r, tn, lane);

    // ---- overlap: out00 = z0l * z0r via direct non-temporal B128 loads (LOADcnt path).
    const v4f* a0 = (const v4f*)z0l + (size_t)t * 32;
    const v4f* b0 = (const v4f*)z0r + (size_t)t * 32;
    v4f p = __builtin_nontemporal_load(a0 + lane) * __builtin_nontemporal_load(b0 + lane);
    v4f* o = (v4f*)out + (size_t)t * 64;         // 256 floats / token
    __builtin_nontemporal_store(p, o + lane);    // out[t, 0:128]

    // ---- current buffer guaranteed resident once only the newest 6 remain in flight.
    wait_async_le6();
    const v4f* la = (const v4f*)(base + (parity ? 2 * Z1_BYTES_PER_TOKEN : 0));
    const v4f* lb = la + Z1_BYTES_PER_TOKEN / 16;
    v4f acc = la[lane] * lb[lane];               // k=0 (reference reduction order)
    acc += la[32 + lane] * lb[32 + lane];        // k=1
    acc += la[64 + lane] * lb[64 + lane];        // k=2
    __builtin_nontemporal_store(acc, o + 32 + lane);  // out[t, 128:256]
    // WAR on the recycled buffer is safe: s_wait_dscnt before consuming la/lb orders
    // this iteration's LDS reads ahead of the async refill issued in the next iteration.
  }
  // Up to 6 dangling (clamped) async copies remain; S_ENDPGM's implicit wait-idle drains them.
}

extern "C" void kernel_launch(void* const* d_in, const int* in_sizes, int n_in,
                              void* d_out, int out_size, void* d_ws, size_t ws_size,
                              hipStream_t stream) {
  const float* z0l = (const float*)d_in[0];  // (TOKENS, 128) f32
  const float* z1l = (const float*)d_in[1];  // (TOKENS, 3, 128) f32
  const float* z0r = (const float*)d_in[2];  // (TOKENS, 128) f32
  const float* z1r = (const float*)d_in[3];  // (TOKENS, 3, 128) f32
  float* out = (float*)d_out;                // (TOKENS, 256) f32

  const int ntokens = in_sizes[0] / RANK;    // 262144

  // 4096 blocks x 8 waves = 32768 waves; 8 tokens per wave via grid-stride.
  etp_o0_kernel<<<dim3(4096), dim3(256), 0, stream>>>(z0l, z1l, z0r, z1r, out, ntokens);
}